// CoordToDist_84335977825047
// MI455X (gfx1250) — compile-verified
//
#include <hip/hip_runtime.h>
#include <math.h>

typedef __attribute__((ext_vector_type(2))) float v2f;
typedef __attribute__((ext_vector_type(8))) float v8f;

#define NPTS   2048
#define NTILE  (NPTS / 16)   // 128 tiles per dimension
#define TJ_PER_WAVE 8        // each wave covers 16 rows x 128 cols

// D[b,i,j] = sqrt(|xi|^2 + |xj|^2 - 2*xi.xj).
// WMMA computes (|xi|^2 - 2*xi.xj) directly via K=4 packing:
//   A row i = ( x_i, y_i, z_i, |x_i|^2 )
//   B col j = (-2x_j, -2y_j, -2z_j, 1 )
// Epilogue adds |x_j|^2, clamps, raw v_sqrt_f32.
__global__ __launch_bounds__(256)
void CoordToDist_wmma_kernel(const float* __restrict__ coords,
                             float* __restrict__ out) {
    const int lane = threadIdx.x & 31;
    const int wid  = (blockIdx.x * blockDim.x + threadIdx.x) >> 5;

    // wid in [0, 8 * 128 * 16): batch, tile-row, tile-col-group (8 tiles)
    const int b    = wid >> 11;          // / (128*16)
    const int rem  = wid & 2047;
    const int ti   = rem >> 4;           // 0..127
    const int tjg  = rem & 15;           // 0..15
    const int i0   = ti << 4;
    const int j0b  = tjg << 7;           // *128

    const float* __restrict__ c = coords + (size_t)b * (NPTS * 3);
    float*       __restrict__ o = out    + (size_t)b * NPTS * NPTS;

    const int half = lane >> 4;          // 0: holds K0,K1 ; 1: holds K2,K3
    const int l15  = lane & 15;

    // ---- A operand: rows i0..i0+15 ----
    // half0 lane: (x, y) ; half1 lane: (z, |xi|^2)
    const int arow = i0 + l15;
    const float ax  = c[arow * 3 + (half ? 2 : 0)];
    const float ayl = c[arow * 3 + 1];
    const float aym = half ? 0.0f : ayl;        // masked y
    const float rpart = ax * ax + aym * aym;    // half0: x^2+y^2 ; half1: z^2
    const float plow  = __shfl(rpart, l15, 32); // x^2+y^2 partial from low half
    v2f a;
    a.x = ax;
    a.y = half ? (rpart + plow) : ayl;          // half1 gets |xi|^2

    // Per-lane constants for B construction: bm.y = fma(ym, -2, addn)
    const float addn = half ? 1.0f : 0.0f;

    // Per-lane base pointers -> all loop accesses become immediate offsets.
    const float* __restrict__ cbx = c + (size_t)(j0b + l15) * 3 + (half ? 2 : 0);
    const float* __restrict__ cby = c + (size_t)(j0b + l15) * 3 + 1;
    float*       __restrict__ ob  = o + (size_t)(i0 + (half << 3)) * NPTS + j0b + l15;

#pragma unroll
    for (int t = 0; t < TJ_PER_WAVE; ++t) {
        // ---- B operand: columns j0b+16t .. +15 ----
        const float bx = cbx[t * 48];            // half0: x_j ; half1: z_j
        const float by = cby[t * 48];            // y_j (both halves load it)
        const float ym = half ? 0.0f : by;
        v2f bm;
        bm.x = -2.0f * bx;
        bm.y = fmaf(ym, -2.0f, addn);            // half0: -2y ; half1: 1

        // Column norm |x_j|^2 broadcast to all lanes of the column.
        const float cpart = bx * bx + ym * ym;   // half0: x^2+y^2 ; half1: z^2
        const float cn = __shfl(cpart, l15, 32) + __shfl(cpart, l15 + 16, 32);

        // G = A x B : G[i][j] = |xi|^2 - 2*xi.xj   (C = 0)
        v8f acc = {};
        acc = __builtin_amdgcn_wmma_f32_16x16x4_f32(
            /*neg_a=*/false, a, /*neg_b=*/false, bm,
            /*c_mod=*/(short)0, acc, /*reuse_a=*/false, /*reuse_b=*/false);

#pragma unroll
        for (int r = 0; r < 8; ++r) {
            const float s = fmaxf(cn + acc[r], 0.0f);
            // Raw v_sqrt_f32 (~1 ULP): inputs are unit-scale, never denormal.
            ob[r * NPTS + (t << 4)] = __builtin_amdgcn_sqrtf(s);
        }
    }
}

extern "C" void kernel_launch(void* const* d_in, const int* in_sizes, int n_in,
                              void* d_out, int out_size, void* d_ws, size_t ws_size,
                              hipStream_t stream) {
    (void)in_sizes; (void)n_in; (void)d_ws; (void)ws_size; (void)out_size;
    const float* coords = (const float*)d_in[0];
    float* out = (float*)d_out;

    // 8 batches * 128 row-tiles * 16 col-groups = 16384 waves; 8 waves/block.
    const int total_waves = 8 * NTILE * (NTILE / TJ_PER_WAVE);
    const int block = 256;
    const int grid  = (total_waves * 32) / block;   // 2048
    CoordToDist_wmma_kernel<<<grid, block, 0, stream>>>(coords, out);
}